// RNNBase_9414568312974
// MI455X (gfx1250) — compile-verified
//
#include <hip/hip_runtime.h>
#include <math.h>
#include <stdint.h>

typedef float v2f __attribute__((ext_vector_type(2)));
typedef float v8f __attribute__((ext_vector_type(8)));

#define RNN_B 32
#define RNN_T 2048
#define RNN_I 256
#define RNN_H 512
#define RNN_O 256
#define SCAN_NBLK 8          // RNN_H / 64 columns per block
#define HSTR 520             // LDS row stride for h buffer (even->b64 ok, %4==0 ->float4 ok)
#define ASTR 20              // LDS row stride for A tile (float4-aligned, conflict-free b64)

__device__ __forceinline__ void wait_asynccnt0() {
#if __has_builtin(__builtin_amdgcn_s_wait_asynccnt)
    __builtin_amdgcn_s_wait_asynccnt(0);
#else
    asm volatile("s_wait_asynccnt 0x0" ::: "memory");
#endif
}

// ---------------------------------------------------------------------------
// Tiled fp32 WMMA GEMM:  C[M x N] = A[M x K] * B[K x N] + bias[N]
// 256 threads / 8 waves, 64x64 tile, K-chunks of 16.
// B tile staged in LDS directly in WMMA B-fragment order -> single ds_load_b64
// per operand in the inner loop.
// ---------------------------------------------------------------------------
__global__ __launch_bounds__(256) void wmma_gemm_bias(
    const float* __restrict__ A, int lda,
    const float* __restrict__ Bw, int ldb,
    const float* __restrict__ bias,
    float* __restrict__ C, int ldc, int K)
{
    __shared__ float As[64 * ASTR];     // natural layout, padded
    __shared__ float Bf[1024];          // frag order: [nt(4)][kk(4)][lane(32)][2]

    const int tid   = threadIdx.x;
    const int lane  = tid & 31;
    const int wave  = tid >> 5;
    const int mt    = wave & 3;
    const int ntb   = (wave >> 2) * 2;
    const int r0    = blockIdx.y * 64;
    const int n0    = blockIdx.x * 64;
    const int l15   = lane & 15;
    const int khalf = (lane >= 16) ? 2 : 0;

    const int arow = tid >> 2;          // 0..63
    const int akq  = (tid & 3) * 4;     // 0,4,8,12
    const float* aG = A + (size_t)(r0 + arow) * lda + akq;

    v8f acc0, acc1;
    const float bz0 = bias[n0 + ntb * 16 + l15];
    const float bz1 = bias[n0 + ntb * 16 + 16 + l15];
#pragma unroll
    for (int v = 0; v < 8; ++v) { acc0[v] = bz0; acc1[v] = bz1; }

    for (int kg = 0; kg < K; kg += 16) {
        // stage A tile 64x16 (one float4 per thread)
        *(float4*)&As[arow * ASTR + akq] = *(const float4*)(aG + kg);
        // stage B tile 16x64 into fragment order (2 slots / thread)
        int s = tid;
#pragma unroll
        for (int q = 0; q < 2; ++q, s += 256) {
            int nt_ = s >> 7, kk_ = (s >> 5) & 3, ln_ = s & 31;
            int k = kg + kk_ * 4 + ((ln_ >= 16) ? 2 : 0);
            int n = n0 + nt_ * 16 + (ln_ & 15);
            Bf[2 * s]     = Bw[(size_t)k * ldb + n];
            Bf[2 * s + 1] = Bw[(size_t)(k + 1) * ldb + n];
        }
        __syncthreads();

        if (kg + 16 < K) {  // prefetch next chunk (global_prefetch_b8)
            __builtin_prefetch(aG + kg + 16, 0, 1);
            __builtin_prefetch(&Bw[(size_t)(kg + 16 + (tid >> 6)) * ldb + n0 + (tid & 63)], 0, 1);
        }

#pragma unroll
        for (int kk = 0; kk < 4; ++kk) {
            v2f a  = *(const v2f*)&As[(mt * 16 + l15) * ASTR + kk * 4 + khalf];
            v2f b0 = *(const v2f*)&Bf[((ntb * 4 + kk) * 32 + lane) * 2];
            v2f b1 = *(const v2f*)&Bf[(((ntb + 1) * 4 + kk) * 32 + lane) * 2];
            acc0 = __builtin_amdgcn_wmma_f32_16x16x4_f32(false, a, false, b0,
                                                         (short)0, acc0, false, false);
            acc1 = __builtin_amdgcn_wmma_f32_16x16x4_f32(false, a, false, b1,
                                                         (short)0, acc1, false, false);
        }
        __syncthreads();
    }

    // C/D layout: VGPR v, lanes 0-15 -> M=v, lanes 16-31 -> M=v+8; N=lane&15
    const int mrow = r0 + mt * 16 + ((lane >= 16) ? 8 : 0);
#pragma unroll
    for (int v = 0; v < 8; ++v) {
        C[(size_t)(mrow + v) * ldc + n0 + ntb * 16 + l15]      = acc0[v];
        C[(size_t)(mrow + v) * ldc + n0 + ntb * 16 + 16 + l15] = acc1[v];
    }
}

// ---------------------------------------------------------------------------
// Persistent recurrent scan. hs (B,T,H) pre-filled with xw = x@W_ih + b_h;
// step t: h_t = tanh(xw_t + h_{t-1} @ W_hh), written in place over xw_t.
// 8 blocks; block j owns H-columns [64j, 64j+64). W_hh slice (128 KB) resident
// in LDS in B-fragment order; h_prev (64 KB) restaged each step via the
// async Global->LDS mover (ASYNCcnt), skipping the VGPR round-trip.
// ---------------------------------------------------------------------------
__global__ __launch_bounds__(256) void rnn_scan(
    const float* __restrict__ h0,
    const float* __restrict__ W_hh,
    float* __restrict__ hs,
    unsigned* __restrict__ bar)
{
    extern __shared__ float smem[];
    float* wfrag = smem;                 // 4*128*32*2 = 32768 floats (128 KB)
    float* hbuf  = smem + 32768;         // 32 x HSTR

    const size_t TH = (size_t)RNN_T * RNN_H;
    const int tid   = threadIdx.x;
    const int lane  = tid & 31;
    const int wave  = tid >> 5;
    const int mt    = wave & 1;
    const int nt    = wave >> 1;
    const int l15   = lane & 15;
    const int khalf = (lane >= 16) ? 2 : 0;
    const int nblk0 = blockIdx.x * 64;

    // One-time: pre-swizzle W_hh[:, nblk0..+64) into B-fragment order.
    for (int s = tid; s < 4 * 128 * 32; s += 256) {
        int nt_ = s >> 12;
        int kk_ = (s >> 5) & 127;
        int ln_ = s & 31;
        int k = kk_ * 4 + ((ln_ >= 16) ? 2 : 0);
        int n = nblk0 + nt_ * 16 + (ln_ & 15);
        wfrag[2 * s]     = W_hh[(size_t)k * RNN_H + n];
        wfrag[2 * s + 1] = W_hh[(size_t)(k + 1) * RNN_H + n];
    }

    // h_prev staging: fixed row per thread -> all row*T*H math hoisted out
    const int srow = tid >> 3;                       // 0..31
    const int c4b  = tid & 7;                        // float4 column base
    const float* hrow_rd = hs + (size_t)srow * TH;   // + (t-1)*H at use
    float* lrow = hbuf + srow * HSTR + c4b * 4;
    const uint32_t lrow_lds = (uint32_t)(uintptr_t)lrow;  // LDS byte address (low 32 bits)

    // compute-side constants
    const int brow = mt * 16 + ((lane >= 16) ? 8 : 0);   // C-layout batch row base
    const int ncol = nblk0 + nt * 16 + l15;
    const float* wbase  = wfrag + nt * 8192 + lane * 2;
    const float* habase = hbuf + (mt * 16 + l15) * HSTR + khalf;
    float* xwbase = hs + (size_t)brow * TH + ncol;

    for (int t = 0; t < RNN_T; ++t) {
        // stage h_prev (32 x 512) into LDS
        if (t == 0) {
#pragma unroll
            for (int j = 0; j < 16; ++j)                      // broadcast h0 over batch
                *(float4*)(lrow + j * 32) = ((const float4*)h0)[c4b + j * 8];
        } else {
            // async Global->LDS b128 copies (GV mode: VDST = LDS addr, VADDR = 64b)
            uint64_t g = (uint64_t)(uintptr_t)(hrow_rd + (size_t)(t - 1) * RNN_H + c4b * 4);
#pragma unroll
            for (int j = 0; j < 16; ++j) {
                asm volatile("global_load_async_to_lds_b128 %0, %1, off"
                             :: "v"(lrow_lds + (uint32_t)(j * 128)),
                                "v"(g + (uint64_t)(j * 128))
                             : "memory");
            }
            wait_asynccnt0();
        }
        __syncthreads();

        // init accumulator with xw_t (in-place slot in hs)
        float* xwt = xwbase + (size_t)t * RNN_H;
        v8f acc;
#pragma unroll
        for (int v = 0; v < 8; ++v) acc[v] = xwt[(size_t)v * TH];

        if (t + 1 < RNN_T) __builtin_prefetch(xwt + RNN_H, 0, 1);

        // K = 512 reduction: 128 x v_wmma_f32_16x16x4_f32 (2x ds_load_b64 each)
#pragma unroll 8
        for (int kk = 0; kk < 128; ++kk) {
            v2f a = *(const v2f*)(habase + kk * 4);
            v2f b = *(const v2f*)(wbase + kk * 64);
            acc = __builtin_amdgcn_wmma_f32_16x16x4_f32(false, a, false, b,
                                                        (short)0, acc, false, false);
        }

        // tanh, write h_t over the xw slot (this IS the hs output)
#pragma unroll
        for (int v = 0; v < 8; ++v)
            xwt[(size_t)v * TH] = tanhf(acc[v]);

        // device-wide barrier (release before, acquire after)
        __threadfence();
        __syncthreads();
        if (tid == 0) {
            unsigned g0 = __hip_atomic_load(&bar[1], __ATOMIC_ACQUIRE, __HIP_MEMORY_SCOPE_AGENT);
            unsigned arrived =
                __hip_atomic_fetch_add(&bar[0], 1u, __ATOMIC_ACQ_REL, __HIP_MEMORY_SCOPE_AGENT);
            if (arrived == SCAN_NBLK - 1) {
                __hip_atomic_store(&bar[0], 0u, __ATOMIC_RELAXED, __HIP_MEMORY_SCOPE_AGENT);
                __hip_atomic_fetch_add(&bar[1], 1u, __ATOMIC_ACQ_REL, __HIP_MEMORY_SCOPE_AGENT);
            } else {
                while (__hip_atomic_load(&bar[1], __ATOMIC_ACQUIRE,
                                         __HIP_MEMORY_SCOPE_AGENT) == g0)
                    __builtin_amdgcn_s_sleep(1);
            }
        }
        __syncthreads();
        __threadfence();   // acquire: invalidate stale lines before next h_prev read
    }
}

__global__ void init_bar(unsigned* bar) { bar[0] = 0u; bar[1] = 0u; }

// ---------------------------------------------------------------------------
extern "C" void kernel_launch(void* const* d_in, const int* in_sizes, int n_in,
                              void* d_out, int out_size, void* d_ws, size_t ws_size,
                              hipStream_t stream) {
    const float* x     = (const float*)d_in[0];
    const float* h0    = (const float*)d_in[1];
    const float* W_ih  = (const float*)d_in[2];
    const float* W_hh  = (const float*)d_in[3];
    const float* b_h   = (const float*)d_in[4];
    const float* W_out = (const float*)d_in[5];
    const float* b_out = (const float*)d_in[6];

    float* os = (float*)d_out;                               // (B,T,O)
    float* hs = os + (size_t)RNN_B * RNN_T * RNN_O;          // (B,T,H), also xw scratch
    unsigned* bar = (unsigned*)d_ws;

    init_bar<<<dim3(1), dim3(1), 0, stream>>>(bar);

    // Phase 1: xw = x @ W_ih + b_h -> hs region (row b*T+t matches (B,T,H))
    wmma_gemm_bias<<<dim3(RNN_H / 64, (RNN_B * RNN_T) / 64), dim3(256), 0, stream>>>(
        x, RNN_I, W_ih, RNN_H, b_h, hs, RNN_H, RNN_I);

    // Phase 2: sequential scan, in place in hs
    const size_t smem_bytes = (size_t)(32768 + 32 * HSTR) * sizeof(float);  // ~193 KB
    rnn_scan<<<dim3(SCAN_NBLK), dim3(256), smem_bytes, stream>>>(h0, W_hh, hs, bar);

    // Phase 3: os = hs @ W_out + b_out
    wmma_gemm_bias<<<dim3(RNN_O / 64, (RNN_B * RNN_T) / 64), dim3(256), 0, stream>>>(
        hs, RNN_H, W_out, RNN_O, b_out, os, RNN_O, RNN_H);
}